// VectorQuantizer_49314814492903
// MI455X (gfx1250) — compile-verified
//
#include <hip/hip_runtime.h>
#include <stdint.h>

typedef float        v2f __attribute__((ext_vector_type(2)));
typedef float        v8f __attribute__((ext_vector_type(8)));
typedef unsigned int v4u __attribute__((ext_vector_type(4)));
typedef int          v4i __attribute__((ext_vector_type(4)));
typedef int          v8i __attribute__((ext_vector_type(8)));

#define CB_N   512
#define NELEM  524288   // 8*16*64*64
#define CHUNK  1024     // elements per (b,h) slab

// ---------------------------------------------------------------------------
// Kernel 1: stable rank-sort of the 512-entry codebook (value + original idx).
// O(N^2) but N=512: ~512 broadcast LDS reads per thread, one barrier total.
// ---------------------------------------------------------------------------
__global__ __launch_bounds__(CB_N) void vq_sort_codebook(
    const float* __restrict__ w, float* __restrict__ sv, int* __restrict__ si) {
  __shared__ float lw[CB_N];
  const int t = threadIdx.x;
  lw[t] = w[t];
  __syncthreads();
  const float v = lw[t];
  int rank = 0;
  for (int i = 0; i < CB_N; ++i) {
    const float u = lw[i];
    rank += (u < v || (u == v && i < t)) ? 1 : 0;   // stable: ties by index
  }
  sv[rank] = v;
  si[rank] = t;
}

// ---------------------------------------------------------------------------
// Kernel 2: main VQ. Block (b,h): TDM-loads contiguous x chunk (padded 17-
// stride layout in LDS), binary-searches nearest codebook entry, writes the
// permuted output coalesced, accumulates block loss partial.
// ---------------------------------------------------------------------------
__global__ __launch_bounds__(256) void vq_main(
    const float* __restrict__ x,
    const float* __restrict__ sv, const int* __restrict__ si,
    float* __restrict__ out, float* __restrict__ partial) {
  __shared__ float s_x[1088];     // 1024 data + 64 pad dwords (1 per 16)
  __shared__ float s_wv[CB_N];
  __shared__ int   s_wi[CB_N];
  __shared__ float s_red[256];

  const int tid = threadIdx.x;
  const int bid = blockIdx.x;               // 0..511 = b*64 + h
  const int b   = bid >> 6;
  const int h   = bid & 63;

  // --- Tensor Data Mover: global x[bid*1024 .. +1024) -> LDS, padded -------
  if (tid < 32) {                            // one wave issues the DMA
    const uint64_t gaddr = (uint64_t)(uintptr_t)(x + (uint64_t)bid * CHUNK);
    const uint32_t laddr = (uint32_t)(uintptr_t)&s_x[0];   // LDS byte offset

    v4u g0;
    g0.x = 1u;                                            // count=1, user mode
    g0.y = laddr;                                         // lds_addr
    g0.z = (uint32_t)(gaddr & 0xFFFFFFFFu);               // global_addr[31:0]
    g0.w = (uint32_t)((gaddr >> 32) & 0x01FFFFFFu)        // global_addr[56:32]
         | (2u << 30);                                    // type = 2 (image)

    v8i g1;
    g1[0] = (int)((2u << 16)        // data_size = 4B
                | (1u << 20)        // pad_enable
                | (3u << 22));      // pad_interval: 16 dwords  (pad_amount=0 -> 1 dword)
    g1[1] = (int)(1024u << 16);     // tensor_dim0 = 1024 (bits [79:48] low half)
    g1[2] = (int)(1u << 16);        // tensor_dim1 = 1
    g1[3] = (int)(1024u << 16);     // tile_dim0   = 1024
    g1[4] = 1;                      // tile_dim1   = 1, tile_dim2 = 0
    g1[5] = 1024;                   // tensor_dim0_stride = 1024
    g1[6] = 0;
    g1[7] = 0;

    const v4i gz4 = {0, 0, 0, 0};               // groups 2/3 unused (2-D tile)
    const v8i gz8 = {0, 0, 0, 0, 0, 0, 0, 0};   // extra group (6-arg variant), unused
    __builtin_amdgcn_tensor_load_to_lds(g0, g1, gz4, gz4, gz8, 0);
  }

  // --- stage sorted codebook into LDS (overlaps with the DMA) --------------
  for (int i = tid; i < CB_N; i += 256) {
    s_wv[i] = sv[i];
    s_wi[i] = si[i];
  }

  if (tid < 32) __builtin_amdgcn_s_wait_tensorcnt(0);
  __syncthreads();

  // --- per-element nearest-neighbor + permuted write + loss ----------------
  float lsum = 0.f;
  const uint64_t outb = (uint64_t)b * 65536u + (uint64_t)h * 64u;
#pragma unroll
  for (int r = 0; r < 4; ++r) {
    const int e = tid + r * 256;            // e = c*64 + w  (output order)
    const int c = e >> 6;
    const int w = e & 63;
    const float xv = s_x[w * 17 + c];       // source element (padded, no conflicts)

    // branchless lower bound: p = min(#{s_wv < xv}, 511)
    int p = 0;
#pragma unroll
    for (int step = 256; step > 0; step >>= 1) {
      const int probe = p + step - 1;       // <= 510, in bounds
      p = (s_wv[probe] < xv) ? (p + step) : p;
    }
    const int il = (p > 0) ? (p - 1) : 0;
    const float wl = s_wv[il];
    const float wr = s_wv[p];
    const float dl = (xv - wl) * (xv - wl); // same fp32 math as reference
    const float dr = (xv - wr) * (xv - wr);
    float q;
    if (dl < dr)      q = wl;
    else if (dr < dl) q = wr;
    else              q = (s_wi[il] < s_wi[p]) ? wl : wr;  // argmin first-index tie

    const uint64_t j = outb + (uint64_t)c * 4096u + (uint64_t)w;
    const float xo = x[j];                  // x at the *output* position (coalesced)
    out[j] = q;                             // quantized_st == quantized numerically
    const float d = xo - q;
    lsum = fmaf(d, d, lsum);
  }

  // --- deterministic block reduction ---------------------------------------
  s_red[tid] = lsum;
  __syncthreads();
  for (int off = 128; off > 0; off >>= 1) {
    if (tid < off) s_red[tid] += s_red[tid + off];
    __syncthreads();
  }
  if (tid == 0) partial[bid] = s_red[0];
}

// ---------------------------------------------------------------------------
// Kernel 3: final loss reduction. Fixed-order lane sums, then a WMMA
// (A = lane partials, B = ones) collapses the 32 lane values; one shfl
// combines the two half-wave row groups. Deterministic.
// ---------------------------------------------------------------------------
__global__ __launch_bounds__(32) void vq_reduce(
    const float* __restrict__ partial, float* __restrict__ out) {
  const int lane = threadIdx.x;             // 32 threads, EXEC all ones for WMMA
  float p = 0.f;
  for (int i = lane; i < 512; i += 32) p += partial[i];   // fixed order

  v2f a;  a.x = p;    a.y = 0.f;            // A[m][k]: only K=0 (lanes 0-15) / K=2 (16-31)
  v2f bm; bm.x = 1.f; bm.y = 1.f;           // B = all-ones -> row sums
  v8f c = {};
  // D[m][n] = p_m + p_{16+m}
  v8f dm = __builtin_amdgcn_wmma_f32_16x16x4_f32(
      false, a, false, bm, (short)0, c, false, false);

  float t = dm[0] + dm[1] + dm[2] + dm[3] + dm[4] + dm[5] + dm[6] + dm[7];
  const float total = t + __shfl(t, lane ^ 16);   // combine M=0..7 / M=8..15 halves

  if (lane == 0) {
    const float m = total * (1.0f / (float)NELEM);
    out[NELEM]     = m;   // q_latent_loss
    out[NELEM + 1] = m;   // e_latent_loss (identical numerically)
  }
}

// ---------------------------------------------------------------------------
extern "C" void kernel_launch(void* const* d_in, const int* in_sizes, int n_in,
                              void* d_out, int out_size, void* d_ws, size_t ws_size,
                              hipStream_t stream) {
  (void)in_sizes; (void)n_in; (void)out_size; (void)ws_size;
  const float* x      = (const float*)d_in[0];   // [8,16,64,64] fp32
  const float* weight = (const float*)d_in[1];   // [1,512] fp32
  float* out = (float*)d_out;                    // 524288 + 2 floats

  float* sv      = (float*)d_ws;                         // 512 sorted values
  int*   si      = (int*)((char*)d_ws + 512 * 4);        // 512 original indices
  float* partial = (float*)((char*)d_ws + 1024 * 4);     // 512 block partials

  vq_sort_codebook<<<1, CB_N, 0, stream>>>(weight, sv, si);
  vq_main<<<512, 256, 0, stream>>>(x, sv, si, out, partial);
  vq_reduce<<<1, 32, 0, stream>>>(partial, out);
}